// GCNPeptideStruct_20461224198768
// MI455X (gfx1250) — compile-verified
//
#include <hip/hip_runtime.h>
#include <hip/hip_bf16.h>

typedef float v2f __attribute__((ext_vector_type(2)));
typedef float v8f __attribute__((ext_vector_type(8)));

#define IN_CH 128
#define HID   128
#define NOUT  11

// Native no-return fp32 global atomic add (GLOBAL_ATOMIC_ADD_F32), device scope.
// Forced via inline asm so the compiler cannot expand to a CAS loop: this is
// the dominant instruction of the whole workload (~620M executions).
static __device__ __forceinline__ void atomic_add_f32(float* p, float v) {
#if defined(__HIP_DEVICE_COMPILE__)
    asm volatile("global_atomic_add_f32 %0, %1, off scope:SCOPE_DEV"
                 :: "v"(p), "v"(v)
                 : "memory");
#else
    __hip_atomic_fetch_add(p, v, __ATOMIC_RELAXED, __HIP_MEMORY_SCOPE_AGENT);
#endif
}

// ---------------------------------------------------------------------------
// GEMM: L[N,128] = H[N,128] @ W[128,128], fp32 WMMA 16x16x4.
// Block = 256 threads = 8 waves; each wave owns one 16-wide column tile,
// block covers 16 rows x 128 cols. W staged in LDS (64 KB).
// ---------------------------------------------------------------------------
__global__ __launch_bounds__(256)
void gcn_gemm_f32_wmma(const float* __restrict__ H, const float* __restrict__ W,
                       float* __restrict__ L, int n_nodes) {
    __shared__ float Ws[128 * 128];
    // cooperative load of W into LDS (float4 wide)
    const float4* Wv = (const float4*)W;
    float4* Wsv = (float4*)Ws;
    for (int i = threadIdx.x; i < (128 * 128) / 4; i += 256) Wsv[i] = Wv[i];
    __syncthreads();

    const int wave = threadIdx.x >> 5;         // 0..7 -> column tile
    const int lane = threadIdx.x & 31;
    const int row0 = blockIdx.x * 16;
    const int col0 = wave * 16;
    const int m     = lane & 15;               // row (A) / col (B,C)
    const int khalf = (lane >> 4) * 2;         // 0 or 2 : K-pair select

    // clamp A-row so every lane issues a valid load (EXEC stays all-ones)
    int arow_i = row0 + m;
    if (arow_i > n_nodes - 1) arow_i = n_nodes - 1;
    const float* __restrict__ arow = H + (size_t)arow_i * 128;

    v8f acc = {};
#pragma unroll
    for (int k0 = 0; k0 < 128; k0 += 4) {
        v2f a, b;
        a.x = arow[k0 + khalf];
        a.y = arow[k0 + khalf + 1];
        b.x = Ws[(k0 + khalf) * 128 + col0 + m];
        b.y = Ws[(k0 + khalf + 1) * 128 + col0 + m];
        acc = __builtin_amdgcn_wmma_f32_16x16x4_f32(
            /*neg_a=*/false, a, /*neg_b=*/false, b,
            /*c_mod=*/(short)0, acc, /*reuse_a=*/false, /*reuse_b=*/false);
    }

    // D layout: VGPR v -> row m = v + (lane>>4)*8, col n = lane&15
    const int rbase = row0 + ((lane >> 4) * 8);
    const int c = col0 + m;
#pragma unroll
    for (int v = 0; v < 8; ++v) {
        const int r = rbase + v;
        if (r < n_nodes) L[(size_t)r * 128 + c] = acc[v];
    }
}

// ---------------------------------------------------------------------------
// Zeroing helpers
// ---------------------------------------------------------------------------
__global__ void zero_f4(float4* __restrict__ p, long long n4) {
    long long t = blockIdx.x * (long long)blockDim.x + threadIdx.x;
    if (t < n4) p[t] = make_float4(0.f, 0.f, 0.f, 0.f);
}
__global__ void zero_f1(float* __restrict__ p, int n) {
    int t = blockIdx.x * blockDim.x + threadIdx.x;
    if (t < n) p[t] = 0.f;
}

// ---------------------------------------------------------------------------
// Degree accumulation + rsqrt (done once; reused for all 3 layers)
// ---------------------------------------------------------------------------
__global__ void deg_accum(const int* __restrict__ col, const float* __restrict__ ew,
                          float* __restrict__ deg, int n_edges) {
    int e = blockIdx.x * blockDim.x + threadIdx.x;
    if (e < n_edges) atomic_add_f32(&deg[col[e]], ew[e]);
}
__global__ void deg_to_dis(float* __restrict__ deg, int n_nodes) {
    int i = blockIdx.x * blockDim.x + threadIdx.x;
    if (i < n_nodes) deg[i] = rsqrtf(deg[i] + 1.0f);   // self-loop adds 1
}

// ---------------------------------------------------------------------------
// Edge scatter: agg[col] += dis[row]*ew*dis[col] * lin[row]
// One thread per (edge, 4-channel quad): 32 threads per edge; the float4
// gather of lin[row] is coalesced, per-edge scalars hit L0.
// ---------------------------------------------------------------------------
__global__ __launch_bounds__(256)
void edge_scatter(const float* __restrict__ lin, const int* __restrict__ row,
                  const int* __restrict__ col, const float* __restrict__ ew,
                  const float* __restrict__ dis, float* __restrict__ agg,
                  int n_edges) {
    long long t = blockIdx.x * (long long)blockDim.x + threadIdx.x;
    int e = (int)(t >> 5);
    if (e >= n_edges) return;
    int cq = ((int)t & 31) * 4;
    int r = row[e], c = col[e];
    float coef = dis[r] * ew[e] * dis[c];
    float4 v = *(const float4*)(lin + (size_t)r * 128 + cq);
    float* dst = agg + (size_t)c * 128 + cq;
    atomic_add_f32(dst + 0, coef * v.x);
    atomic_add_f32(dst + 1, coef * v.y);
    atomic_add_f32(dst + 2, coef * v.z);
    atomic_add_f32(dst + 3, coef * v.w);
}

// ---------------------------------------------------------------------------
// Finalize: h = [relu](agg + dis^2*lin + bias)
// ---------------------------------------------------------------------------
__global__ __launch_bounds__(256)
void layer_finalize(const float* __restrict__ agg, const float* __restrict__ lin,
                    const float* __restrict__ dis, const float* __restrict__ bias,
                    float* __restrict__ hout, int n_nodes, int do_relu) {
    long long t = blockIdx.x * (long long)blockDim.x + threadIdx.x;
    int node = (int)(t >> 5);
    if (node >= n_nodes) return;
    int cq = ((int)t & 31) * 4;
    float d = dis[node];
    float s = d * d;
    float4 a = *(const float4*)(agg + (size_t)node * 128 + cq);
    float4 l = *(const float4*)(lin + (size_t)node * 128 + cq);
    float4 b = *(const float4*)(bias + cq);
    float4 r;
    r.x = a.x + s * l.x + b.x;
    r.y = a.y + s * l.y + b.y;
    r.z = a.z + s * l.z + b.z;
    r.w = a.w + s * l.w + b.w;
    if (do_relu) {
        r.x = fmaxf(r.x, 0.f); r.y = fmaxf(r.y, 0.f);
        r.z = fmaxf(r.z, 0.f); r.w = fmaxf(r.w, 0.f);
    }
    *(float4*)(hout + (size_t)node * 128 + cq) = r;
}

// ---------------------------------------------------------------------------
// Pooling: per-graph sums and counts
// ---------------------------------------------------------------------------
__global__ void pool_count(const int* __restrict__ batch, float* __restrict__ cnt,
                           int n_nodes) {
    int i = blockIdx.x * blockDim.x + threadIdx.x;
    if (i < n_nodes) atomic_add_f32(&cnt[batch[i]], 1.0f);
}
__global__ __launch_bounds__(256)
void pool_sum(const float* __restrict__ h, const int* __restrict__ batch,
              float* __restrict__ sums, int n_nodes) {
    long long t = blockIdx.x * (long long)blockDim.x + threadIdx.x;
    int node = (int)(t >> 5);
    if (node >= n_nodes) return;
    int cq = ((int)t & 31) * 4;
    int g = batch[node];
    float4 v = *(const float4*)(h + (size_t)node * 128 + cq);
    float* dst = sums + (size_t)g * 128 + cq;
    atomic_add_f32(dst + 0, v.x);
    atomic_add_f32(dst + 1, v.y);
    atomic_add_f32(dst + 2, v.z);
    atomic_add_f32(dst + 3, v.w);
}

// ---------------------------------------------------------------------------
// Final FC: out[g, :11] = (sums[g]/max(cnt,1)) @ fcW + fcb
// ---------------------------------------------------------------------------
__global__ void final_fc(const float* __restrict__ sums, const float* __restrict__ cnt,
                         const float* __restrict__ fcW, const float* __restrict__ fcb,
                         float* __restrict__ out, int n_graphs) {
    int g = blockIdx.x * blockDim.x + threadIdx.x;
    if (g >= n_graphs) return;
    float inv = 1.0f / fmaxf(cnt[g], 1.0f);
    float acc[NOUT];
#pragma unroll
    for (int o = 0; o < NOUT; ++o) acc[o] = fcb[o];
    for (int c = 0; c < HID; ++c) {
        float p = sums[(size_t)g * HID + c] * inv;
#pragma unroll
        for (int o = 0; o < NOUT; ++o) acc[o] += p * fcW[c * NOUT + o];
    }
#pragma unroll
    for (int o = 0; o < NOUT; ++o) out[(size_t)g * NOUT + o] = acc[o];
}

// ---------------------------------------------------------------------------
// Host orchestration
// ---------------------------------------------------------------------------
extern "C" void kernel_launch(void* const* d_in, const int* in_sizes, int n_in,
                              void* d_out, int out_size, void* d_ws, size_t ws_size,
                              hipStream_t stream) {
    const float* x    = (const float*)d_in[0];
    const int*   eidx = (const int*)d_in[1];
    const int*   batch= (const int*)d_in[2];
    const float* ew   = (const float*)d_in[3];
    const float* W1 = (const float*)d_in[4];  const float* b1 = (const float*)d_in[5];
    const float* W2 = (const float*)d_in[6];  const float* b2 = (const float*)d_in[7];
    const float* W3 = (const float*)d_in[8];  const float* b3 = (const float*)d_in[9];
    const float* fcW = (const float*)d_in[10]; const float* fcb = (const float*)d_in[11];
    float* out = (float*)d_out;

    const int n_edges  = in_sizes[1] / 2;
    const int n_nodes  = in_sizes[2];
    const int n_graphs = out_size / NOUT;
    const int* rowi = eidx;
    const int* coli = eidx + n_edges;

    // workspace layout (floats)
    float* h    = (float*)d_ws;                         // n_nodes*128
    float* lin  = h   + (size_t)n_nodes * HID;          // n_nodes*128
    float* agg  = lin + (size_t)n_nodes * HID;          // n_nodes*128
    float* dis  = agg + (size_t)n_nodes * HID;          // n_nodes  (deg -> rsqrt in place)
    float* sums = dis + n_nodes;                        // n_graphs*128
    float* cnt  = sums + (size_t)n_graphs * HID;        // n_graphs

    const long long nh4   = (long long)n_nodes * HID / 4;
    const int blkNH4      = (int)((nh4 + 255) / 256);
    const long long nEthr = (long long)n_edges * 32;
    const int blkE        = (int)((nEthr + 255) / 256);
    const long long nNthr = (long long)n_nodes * 32;
    const int blkN32      = (int)((nNthr + 255) / 256);
    const int blkN        = (n_nodes + 255) / 256;
    const int blkEdge     = (n_edges + 255) / 256;
    const int rowTiles    = (n_nodes + 15) / 16;

    // degrees -> dis (once, shared by all layers)
    zero_f1<<<blkN, 256, 0, stream>>>(dis, n_nodes);
    deg_accum<<<blkEdge, 256, 0, stream>>>(coli, ew, dis, n_edges);
    deg_to_dis<<<blkN, 256, 0, stream>>>(dis, n_nodes);

    const float* layer_in[3] = { x, h, h };
    const float* Ws[3]  = { W1, W2, W3 };
    const float* bs[3]  = { b1, b2, b3 };
    for (int L = 0; L < 3; ++L) {
        gcn_gemm_f32_wmma<<<rowTiles, 256, 0, stream>>>(layer_in[L], Ws[L], lin, n_nodes);
        zero_f4<<<blkNH4, 256, 0, stream>>>((float4*)agg, nh4);
        edge_scatter<<<blkE, 256, 0, stream>>>(lin, rowi, coli, ew, dis, agg, n_edges);
        layer_finalize<<<blkN32, 256, 0, stream>>>(agg, lin, dis, bs[L], h, n_nodes, 1);
    }

    // global mean pool + FC
    const long long ng4 = (long long)n_graphs * HID / 4;
    zero_f4<<<(int)((ng4 + 255) / 256), 256, 0, stream>>>((float4*)sums, ng4);
    zero_f1<<<(n_graphs + 255) / 256, 256, 0, stream>>>(cnt, n_graphs);
    pool_count<<<blkN, 256, 0, stream>>>(batch, cnt, n_nodes);
    pool_sum<<<blkN32, 256, 0, stream>>>(h, batch, sums, n_nodes);
    final_fc<<<(n_graphs + 255) / 256, 256, 0, stream>>>(sums, cnt, fcW, fcb, out, n_graphs);
}